// Model_36867999269135
// MI455X (gfx1250) — compile-verified
//
#include <hip/hip_runtime.h>
#include <hip/hip_bf16.h>

typedef __attribute__((ext_vector_type(2))) float v2f;
typedef __attribute__((ext_vector_type(8))) float v8f;

#define CONTEXT 7

struct Params {
  const float *w0,*b0,*w1,*b1,*w2,*b2,*w3,*b3,*w4,*b4,*w5,*b5,*w6,*b6,
              *w7,*b7,*w8,*b8,*w9,*b9,*wfc,*bfc;
  const float *mc0w,*mc0b,*bn0g,*bn0b,*bn0m,*bn0v;
  const float *mc1w,*mc1b,*bn1g,*bn1b,*bn1m,*bn1v;
  const float *mc2w,*mc2b,*bn2g,*bn2b,*bn2m,*bn2v;
  const float *mc3w,*mc3b,*bn3g,*bn3b,*bn3m,*bn3v;
  const float *mc4w,*mc4b,*bn4g,*bn4b,*bn4m,*bn4v;
  const float *mfw,*mfb;
};
static_assert(sizeof(Params) == 54 * sizeof(const float*), "params layout");

__device__ __forceinline__ v8f wmma4(v2f a, v2f b, v8f c) {
  // D = A(16x4,f32) * B(4x16,f32) + C ; exact fp32 matrix op on CDNA5
  return __builtin_amdgcn_wmma_f32_16x16x4_f32(false, a, false, b, (short)0, c,
                                               false, false);
}

__device__ __forceinline__ float relu(float a) { return a > 0.f ? a : 0.f; }

// ---- LDS layout (floats), overlapped across pipeline stages -----------------
#define OFF_H1   0        // [8][12][180]  = 17280  (stage A out, read by c3)
#define OFF_H0   17280    // [38][192]     =  7296  (padded pooled input)
#define OFF_C0   24576    // [16][3][192]  =  9216  (c0 rows d-1..d+1)
#define OFF_A1   33792    // [16][48]      =   768  (c1 weights, M-padded)
#define OFF_C1R  34560    // [8][192]      =  1536  (c1 row output)
#define OFF_H3   17280    // [8][12][130]  = 12480  (c3+c4 out)   reuses H0..
#define OFF_A5   29760    // [16][72]      =  1152
#define OFF_H5   0        // [16][12][88]  = 16896  reuses H1 (dead)
#define OFF_A6   29760    // [16][144]     =  2304  reuses A5 (dead)
#define OFF_H6   17280    // [16][12][54]  = 10368  reuses H3 (dead)
#define OFF_H7   0        // [8][12][42]   =  4032  reuses H5 (dead)
#define OFF_A8   4032     // [16][72]      =  1152
#define OFF_H8   5184     // [16][12][20]  =  3840
#define OFF_H9   9024     // [16][12][8]   =  1536
#define OFF_FC   10560    // [16][2]
#define OFF_MA   10592    // [32][2]
#define OFF_MB   10656    // [32][2]
#define SMEM_F   36096    // 144,384 bytes < 320KB WGP LDS; 2 blocks/WGP

__global__ __launch_bounds__(256)
void fused_harmonic_net(const float* __restrict__ x, float* __restrict__ out,
                        Params P) {
  __shared__ float sm[SMEM_F];
  const int tid  = threadIdx.x;
  const int lane = tid & 31;
  const int wv   = tid >> 5;        // wave 0..7 (wave32)
  const int half = lane >> 4;
  const int l16  = lane & 15;
  const int s = blockIdx.x;         // sample = b*128 + t
  const int b = s >> 7;
  const int t = s & 127;

  // ---- Stage 0: windowed input + avgpool6 over D -> H0 padded [38][192] ----
  for (int i = tid; i < 38 * 192; i += 256) {
    int r = i / 192, pos = i % 192;
    float v = 0.f;
    if (r >= 1 && r <= 36 && pos < 180) {
      int d = r - 1, win = pos / 12, h = pos % 12;
      int u = t + win - CONTEXT;
      if (u >= 0 && u < 128) {
        const float* xr = x + (size_t)(b * 128 + u) * 2592 + d * 72 + h;
        float a = 0.f;
#pragma unroll
        for (int k = 0; k < 6; ++k) a += xr[k * 12];
        v = a * (1.f / 6.f);
      }
    }
    sm[OFF_H0 + i] = v;
  }
  // c1 A-matrix, 8 out-ch padded to 16 rows: A[oc][ic*3+kt]
  for (int i = tid; i < 16 * 48; i += 256) {
    float v = 0.f;
    if (i < 8 * 48) v = P.w1[i];
    sm[OFF_A1 + i] = v;
  }
  __syncthreads();

  // ---- Stage A: c0 -> c1(WMMA) -> c2 -> avgpool3 over D -> H1[8][12][180] --
  for (int dp = 0; dp < 12; ++dp) {
    for (int dd = 0; dd < 3; ++dd) {
      const int d = dp * 3 + dd;
      // c0 rows d-1..d+1 (zero-padded in D), relu, into C0[16][3][192]
      for (int i = tid; i < 16 * 3 * 192; i += 256) {
        int oc = i / 576, rem = i % 576, slot = rem / 192, pos = rem % 192;
        float v = 0.f;
        int r = d - 1 + slot;
        if (r >= 0 && r <= 35 && pos < 180) {
          float a = P.b0[oc];
#pragma unroll
          for (int kt = 0; kt < 3; ++kt)
            a += P.w0[oc * 3 + kt] * sm[OFF_H0 + (r + kt) * 192 + pos];
          v = relu(a);
        }
        sm[OFF_C0 + i] = v;
      }
      __syncthreads();
      // c1 row d: [16pad x 48] x [48 x 192] via WMMA f32 16x16x4
      for (int task = wv; task < 12; task += 8) {
        const int n0 = task * 16;
        v8f acc = {0.f, 0.f, 0.f, 0.f, 0.f, 0.f, 0.f, 0.f};
#pragma unroll
        for (int ks = 0; ks < 12; ++ks) {
          const int k = ks * 4 + 2 * half;
          v2f a, bb;
          a.x  = sm[OFF_A1 + l16 * 48 + k];
          a.y  = sm[OFF_A1 + l16 * 48 + k + 1];
          bb.x = sm[OFF_C0 + k * 192 + n0 + l16];
          bb.y = sm[OFF_C0 + (k + 1) * 192 + n0 + l16];
          acc = wmma4(a, bb, acc);
        }
        if (half == 0) {      // only M=0..7 are real out-channels
#pragma unroll
          for (int g = 0; g < 8; ++g)
            sm[OFF_C1R + g * 192 + n0 + l16] = relu(acc[g] + P.b1[g]);
        }
      }
      __syncthreads();
      // c2 (8x8 pointwise) + relu + avgpool over D(3) accumulation
      if (tid < 180) {
        float c1v[8];
#pragma unroll
        for (int ic = 0; ic < 8; ++ic) c1v[ic] = sm[OFF_C1R + ic * 192 + tid];
#pragma unroll
        for (int o = 0; o < 8; ++o) {
          float a = P.b2[o];
#pragma unroll
          for (int ic = 0; ic < 8; ++ic) a += P.w2[o * 8 + ic] * c1v[ic];
          a = relu(a) * (1.f / 3.f);
          int idx = OFF_H1 + (o * 12 + dp) * 180 + tid;
          if (dd == 0) sm[idx] = a; else sm[idx] += a;
        }
      }
      __syncthreads();
    }
  }

  // ---- c3 (8,8,3x3) + c4 (8,8,1x1) fused -> H3[8][12][130] ----------------
  for (int i = tid; i < 12 * 130; i += 256) {
    int d = i / 130, rem = i % 130, y = rem / 10, xw = rem % 10;
    float c3v[8];
#pragma unroll
    for (int o = 0; o < 8; ++o) {
      float a = P.b3[o];
      for (int ic = 0; ic < 8; ++ic) {
        const float* hp = &sm[OFF_H1 + (ic * 12 + d) * 180 + y * 12 + xw];
        const float* wp = &P.w3[(o * 8 + ic) * 9];
#pragma unroll
        for (int dy = 0; dy < 3; ++dy)
#pragma unroll
          for (int dx = 0; dx < 3; ++dx) a += wp[dy * 3 + dx] * hp[dy * 12 + dx];
      }
      c3v[o] = relu(a);
    }
#pragma unroll
    for (int o = 0; o < 8; ++o) {
      float a = P.b4[o];
#pragma unroll
      for (int ic = 0; ic < 8; ++ic) a += P.w4[o * 8 + ic] * c3v[ic];
      sm[OFF_H3 + (o * 12 + d) * 130 + rem] = relu(a);
    }
  }
  __syncthreads();

  // ---- c5 (16,8,3x3) via WMMA: K=72, N=11*8=88 per d -> H5[16][12][88] ----
  for (int i = tid; i < 16 * 72; i += 256) sm[OFF_A5 + i] = P.w5[i];
  __syncthreads();
  for (int task = wv; task < 12 * 6; task += 8) {
    const int d = task / 6, n0 = (task % 6) * 16;
    const int n = n0 + l16, y = n / 8, xw = n % 8;
    v8f acc = {0.f, 0.f, 0.f, 0.f, 0.f, 0.f, 0.f, 0.f};
    for (int ks = 0; ks < 18; ++ks) {
      const int k = ks * 4 + 2 * half;
      v2f a, bb;
      a.x = sm[OFF_A5 + l16 * 72 + k];
      a.y = sm[OFF_A5 + l16 * 72 + k + 1];
      float b0v = 0.f, b1v = 0.f;
      if (n < 88) {
        int i0 = k / 9, j0 = k % 9, i1 = (k + 1) / 9, j1 = (k + 1) % 9;
        b0v = sm[OFF_H3 + (i0 * 12 + d) * 130 + (y + j0 / 3) * 10 + (xw + j0 % 3)];
        b1v = sm[OFF_H3 + (i1 * 12 + d) * 130 + (y + j1 / 3) * 10 + (xw + j1 % 3)];
      }
      bb.x = b0v; bb.y = b1v;
      acc = wmma4(a, bb, acc);
    }
    if (n < 88) {
#pragma unroll
      for (int g = 0; g < 8; ++g) {
        int m = half * 8 + g;
        sm[OFF_H5 + (m * 12 + d) * 88 + n] = relu(acc[g] + P.b5[m]);
      }
    }
  }
  __syncthreads();

  // ---- c6 (16,16,3x3) via WMMA: K=144, N=9*6=54 per d -> H6[16][12][54] ---
  for (int i = tid; i < 16 * 144; i += 256) sm[OFF_A6 + i] = P.w6[i];
  __syncthreads();
  for (int task = wv; task < 12 * 4; task += 8) {
    const int d = task / 4, n0 = (task % 4) * 16;
    const int n = n0 + l16, y = n / 6, xw = n % 6;
    v8f acc = {0.f, 0.f, 0.f, 0.f, 0.f, 0.f, 0.f, 0.f};
    for (int ks = 0; ks < 36; ++ks) {
      const int k = ks * 4 + 2 * half;
      v2f a, bb;
      a.x = sm[OFF_A6 + l16 * 144 + k];
      a.y = sm[OFF_A6 + l16 * 144 + k + 1];
      float b0v = 0.f, b1v = 0.f;
      if (n < 54) {
        int i0 = k / 9, j0 = k % 9, i1 = (k + 1) / 9, j1 = (k + 1) % 9;
        b0v = sm[OFF_H5 + (i0 * 12 + d) * 88 + (y + j0 / 3) * 8 + (xw + j0 % 3)];
        b1v = sm[OFF_H5 + (i1 * 12 + d) * 88 + (y + j1 / 3) * 8 + (xw + j1 % 3)];
      }
      bb.x = b0v; bb.y = b1v;
      acc = wmma4(a, bb, acc);
    }
    if (n < 54) {
#pragma unroll
      for (int g = 0; g < 8; ++g) {
        int m = half * 8 + g;
        sm[OFF_H6 + (m * 12 + d) * 54 + n] = relu(acc[g] + P.b6[m]);
      }
    }
  }
  __syncthreads();

  // ---- avgpool (1,3,1) stride 1 + c7 (8,16,1x1) -> H7[8][12][42] ----------
  for (int i = tid; i < 12 * 42; i += 256) {
    int d = i / 42, rem = i % 42, y = rem / 6, xw = rem % 6;
    float pv[16];
#pragma unroll
    for (int ic = 0; ic < 16; ++ic) {
      const float* hp = &sm[OFF_H6 + (ic * 12 + d) * 54 + y * 6 + xw];
      pv[ic] = (hp[0] + hp[6] + hp[12]) * (1.f / 3.f);
    }
#pragma unroll
    for (int o = 0; o < 8; ++o) {
      float a = P.b7[o];
#pragma unroll
      for (int ic = 0; ic < 16; ++ic) a += P.w7[o * 16 + ic] * pv[ic];
      sm[OFF_H7 + (o * 12 + d) * 42 + rem] = relu(a);
    }
  }
  __syncthreads();

  // ---- c8 (16,8,3x3) via WMMA: K=72, N=5*4=20 per d -> H8[16][12][20] -----
  for (int i = tid; i < 16 * 72; i += 256) sm[OFF_A8 + i] = P.w8[i];
  __syncthreads();
  for (int task = wv; task < 12 * 2; task += 8) {
    const int d = task / 2, n0 = (task % 2) * 16;
    const int n = n0 + l16, y = n / 4, xw = n % 4;
    v8f acc = {0.f, 0.f, 0.f, 0.f, 0.f, 0.f, 0.f, 0.f};
    for (int ks = 0; ks < 18; ++ks) {
      const int k = ks * 4 + 2 * half;
      v2f a, bb;
      a.x = sm[OFF_A8 + l16 * 72 + k];
      a.y = sm[OFF_A8 + l16 * 72 + k + 1];
      float b0v = 0.f, b1v = 0.f;
      if (n < 20) {
        int i0 = k / 9, j0 = k % 9, i1 = (k + 1) / 9, j1 = (k + 1) % 9;
        b0v = sm[OFF_H7 + (i0 * 12 + d) * 42 + (y + j0 / 3) * 6 + (xw + j0 % 3)];
        b1v = sm[OFF_H7 + (i1 * 12 + d) * 42 + (y + j1 / 3) * 6 + (xw + j1 % 3)];
      }
      bb.x = b0v; bb.y = b1v;
      acc = wmma4(a, bb, acc);
    }
    if (n < 20) {
#pragma unroll
      for (int g = 0; g < 8; ++g) {
        int m = half * 8 + g;
        sm[OFF_H8 + (m * 12 + d) * 20 + n] = relu(acc[g] + P.b8[m]);
      }
    }
  }
  __syncthreads();

  // ---- c9 (16,16,(4,1) over H), no relu -> H9[16][12][2][4] ---------------
  for (int i = tid; i < 16 * 96; i += 256) {
    int o = i / 96, rem = i % 96, d = rem / 8, p8 = rem % 8;
    int y = p8 / 4, xw = p8 % 4;
    float a = P.b9[o];
    for (int ic = 0; ic < 16; ++ic)
#pragma unroll
      for (int ky = 0; ky < 4; ++ky)
        a += P.w9[(o * 16 + ic) * 4 + ky] *
             sm[OFF_H8 + (ic * 12 + d) * 20 + (y + ky) * 4 + xw];
    sm[OFF_H9 + i] = a;
  }
  __syncthreads();

  // ---- b5_fc: contract (ic=16, D=12, W=4) -> FC[16][2] --------------------
  if (tid < 32) {
    int o = tid >> 1, yy = tid & 1;
    float a = P.bfc[o];
    for (int ic = 0; ic < 16; ++ic)
      for (int dk = 0; dk < 12; ++dk)
#pragma unroll
        for (int wk = 0; wk < 4; ++wk)
          a += P.wfc[((o * 16 + ic) * 12 + dk) * 4 + wk] *
               sm[OFF_H9 + (ic * 12 + dk) * 8 + yy * 4 + wk];
    sm[OFF_FC + tid] = a;
  }
  __syncthreads();

  // ---- 1D head over length-2 axis -----------------------------------------
  // m_c0 (16,16,3), pad(1,1), bn0 -> MA[16][2]
  if (tid < 32) {
    int c = tid >> 1, j = tid & 1;
    float a = P.mc0b[c];
    for (int ic = 0; ic < 16; ++ic)
#pragma unroll
      for (int kt = 0; kt < 3; ++kt) {
        int ip = j + kt;   // padded index, data at 1,2
        float mv = (ip == 1) ? sm[OFF_FC + ic * 2] :
                   (ip == 2) ? sm[OFF_FC + ic * 2 + 1] : 0.f;
        a += P.mc0w[(c * 16 + ic) * 3 + kt] * mv;
      }
    float inv = P.bn0g[c] * rsqrtf(P.bn0v[c] + 1e-5f);
    sm[OFF_MA + tid] = inv * (a - P.bn0m[c]) + P.bn0b[c];
  }
  __syncthreads();
  // m_c1 (8,16,1), bn1 -> MB[8][2]
  if (tid < 16) {
    int c = tid >> 1, j = tid & 1;
    float a = P.mc1b[c];
    for (int ic = 0; ic < 16; ++ic) a += P.mc1w[c * 16 + ic] * sm[OFF_MA + ic * 2 + j];
    float inv = P.bn1g[c] * rsqrtf(P.bn1v[c] + 1e-5f);
    sm[OFF_MB + tid] = inv * (a - P.bn1m[c]) + P.bn1b[c];
  }
  __syncthreads();
  // m_c2 (16,8,12), pad(5,6), bn2 -> MA[16][2]
  if (tid < 32) {
    int c = tid >> 1, j = tid & 1;
    float a = P.mc2b[c];
    for (int ic = 0; ic < 8; ++ic)
      for (int kt = 0; kt < 12; ++kt) {
        int ip = j + kt;   // data at 5,6
        float mv = (ip == 5) ? sm[OFF_MB + ic * 2] :
                   (ip == 6) ? sm[OFF_MB + ic * 2 + 1] : 0.f;
        a += P.mc2w[(c * 8 + ic) * 12 + kt] * mv;
      }
    float inv = P.bn2g[c] * rsqrtf(P.bn2v[c] + 1e-5f);
    sm[OFF_MA + tid] = inv * (a - P.bn2m[c]) + P.bn2b[c];
  }
  __syncthreads();
  // m_c3 (8,16,1), bn3 -> MB[8][2]
  if (tid < 16) {
    int c = tid >> 1, j = tid & 1;
    float a = P.mc3b[c];
    for (int ic = 0; ic < 16; ++ic) a += P.mc3w[c * 16 + ic] * sm[OFF_MA + ic * 2 + j];
    float inv = P.bn3g[c] * rsqrtf(P.bn3v[c] + 1e-5f);
    sm[OFF_MB + tid] = inv * (a - P.bn3m[c]) + P.bn3b[c];
  }
  __syncthreads();
  // m_c4 (32,8,12), pad(5,6), bn4 -> MA[32][2]
  if (tid < 64) {
    int c = tid >> 1, j = tid & 1;
    float a = P.mc4b[c];
    for (int ic = 0; ic < 8; ++ic)
      for (int kt = 0; kt < 12; ++kt) {
        int ip = j + kt;
        float mv = (ip == 5) ? sm[OFF_MB + ic * 2] :
                   (ip == 6) ? sm[OFF_MB + ic * 2 + 1] : 0.f;
        a += P.mc4w[(c * 8 + ic) * 12 + kt] * mv;
      }
    float inv = P.bn4g[c] * rsqrtf(P.bn4v[c] + 1e-5f);
    sm[OFF_MA + tid] = inv * (a - P.bn4m[c]) + P.bn4b[c];
  }
  __syncthreads();
  // m_fc (25,32,1) + mean over the 2 positions (pointwise => mean commutes)
  if (tid < 25) {
    float a = P.mfb[tid];
    for (int ic = 0; ic < 32; ++ic)
      a += P.mfw[tid * 32 + ic] * 0.5f * (sm[OFF_MA + ic * 2] + sm[OFF_MA + ic * 2 + 1]);
    out[(size_t)s * 25 + tid] = a;
  }
}

extern "C" void kernel_launch(void* const* d_in, const int* in_sizes, int n_in,
                              void* d_out, int out_size, void* d_ws, size_t ws_size,
                              hipStream_t stream) {
  (void)in_sizes; (void)n_in; (void)d_ws; (void)ws_size; (void)out_size;
  const float* x = (const float*)d_in[0];
  Params P;
  const float** fp = (const float**)&P;
  for (int k = 0; k < 54; ++k) fp[k] = (const float*)d_in[1 + k];
  fused_harmonic_net<<<1024, 256, 0, stream>>>(x, (float*)d_out, P);
}